// IFNeuron_10557029614158
// MI455X (gfx1250) — compile-verified
//
#include <hip/hip_runtime.h>

#ifndef __has_builtin
#define __has_builtin(x) 0
#endif

#define IF_T 10
#define IF_ALPHA 0.5f
#define IF_BOUND 0.0005f
#define IF_BLOCK 256

typedef float vf4 __attribute__((ext_vector_type(4)));

// Vectorized main kernel: one float4 column per thread, scan over T in
// registers. On gfx1250 the 10 timestep tiles are staged with
// global_load_async_to_lds_b128 (ASYNCcnt, no VGPR destination) and drained
// in order with s_wait_asynccnt; spikes leave via non-temporal 128-bit stores
// so the L2-resident input is not evicted between graph replays.
__global__ __launch_bounds__(IF_BLOCK) void if_scan_v4(
    const float* __restrict__ x, float* __restrict__ out,
    const float* __restrict__ vthp, unsigned n4) {
  const unsigned tid = threadIdx.x;
  const unsigned gid = blockIdx.x * (unsigned)IF_BLOCK + tid;

  // Vth clamp: relu(Vth - bound) + bound, exact fp32 order as reference.
  const float vth = fmaxf(vthp[0] - IF_BOUND, 0.0f) + IF_BOUND;
  const float thr = IF_ALPHA * vth;

  if (gid >= n4) return;

  vf4* __restrict__ ov = (vf4*)out;
  vf4 mem = {0.0f, 0.0f, 0.0f, 0.0f};

#if defined(__gfx1250__)
  // ---- CDNA5 async-to-LDS pipeline: 10 tiles in flight per wave ----
  __shared__ float lbuf[IF_T][IF_BLOCK * 4];
  const char* gbase = (const char*)x + (size_t)gid * 16u;
  const size_t tstride = (size_t)n4 * 16u;  // bytes between timesteps

#pragma unroll
  for (int t = 0; t < IF_T; ++t) {
    // AS3 pointer value == LDS byte offset expected in the VDST VGPR.
    const unsigned lds_off = (unsigned)(size_t)(
        __attribute__((address_space(3))) void*)(&lbuf[t][tid * 4u]);
    const unsigned long long gaddr =
        (unsigned long long)(gbase + (size_t)t * tstride);
    asm volatile("global_load_async_to_lds_b128 %0, %1, off"
                 :
                 : "v"(lds_off), "v"(gaddr)
                 : "memory");
  }

#pragma unroll
  for (int t = 0; t < IF_T; ++t) {
    // Async loads complete in order: <= (9-t) outstanding => tile t is in LDS.
    asm volatile("s_wait_asynccnt %0" ::"i"(IF_T - 1 - t) : "memory");
    const vf4 xi = *(const vf4*)(&lbuf[t][tid * 4u]);

    mem.x += xi.x; mem.y += xi.y; mem.z += xi.z; mem.w += xi.w;
    vf4 sp;
    sp.x = (mem.x >= thr) ? vth : 0.0f;
    sp.y = (mem.y >= thr) ? vth : 0.0f;
    sp.z = (mem.z >= thr) ? vth : 0.0f;
    sp.w = (mem.w >= thr) ? vth : 0.0f;
    mem.x -= sp.x; mem.y -= sp.y; mem.z -= sp.z; mem.w -= sp.w;

    __builtin_nontemporal_store(sp, &ov[(size_t)t * n4 + gid]);
  }
#else
  // ---- Fallback (host pass / non-gfx1250): 10-deep unrolled direct loads ----
  const vf4* __restrict__ xv = (const vf4*)x;
#pragma unroll
  for (int t = 0; t < IF_T; ++t) {
    const vf4 xi = xv[(size_t)t * n4 + gid];
    mem.x += xi.x; mem.y += xi.y; mem.z += xi.z; mem.w += xi.w;
    vf4 sp;
    sp.x = (mem.x >= thr) ? vth : 0.0f;
    sp.y = (mem.y >= thr) ? vth : 0.0f;
    sp.z = (mem.z >= thr) ? vth : 0.0f;
    sp.w = (mem.w >= thr) ? vth : 0.0f;
    mem.x -= sp.x; mem.y -= sp.y; mem.z -= sp.z; mem.w -= sp.w;
    __builtin_nontemporal_store(sp, &ov[(size_t)t * n4 + gid]);
  }
#endif
}

// Scalar safety net for N not divisible by 4 (not hit for the given shape).
__global__ __launch_bounds__(IF_BLOCK) void if_scan_scalar(
    const float* __restrict__ x, float* __restrict__ out,
    const float* __restrict__ vthp, unsigned n) {
  const unsigned i = blockIdx.x * (unsigned)IF_BLOCK + threadIdx.x;
  const float vth = fmaxf(vthp[0] - IF_BOUND, 0.0f) + IF_BOUND;
  const float thr = IF_ALPHA * vth;
  if (i >= n) return;
  float mem = 0.0f;
#pragma unroll
  for (int t = 0; t < IF_T; ++t) {
    mem += x[(size_t)t * n + i];
    const float s = (mem >= thr) ? vth : 0.0f;
    mem -= s;
    __builtin_nontemporal_store(s, &out[(size_t)t * n + i]);
  }
}

extern "C" void kernel_launch(void* const* d_in, const int* in_sizes, int n_in,
                              void* d_out, int out_size, void* d_ws,
                              size_t ws_size, hipStream_t stream) {
  (void)n_in; (void)out_size; (void)d_ws; (void)ws_size;
  const float* x = (const float*)d_in[0];
  const float* vth = (const float*)d_in[1];
  float* out = (float*)d_out;

  const size_t total = (size_t)in_sizes[0];
  const unsigned n = (unsigned)(total / IF_T);  // elements per timestep

  if ((n & 3u) == 0u) {
    const unsigned n4 = n >> 2;
    const unsigned blocks = (n4 + IF_BLOCK - 1u) / IF_BLOCK;
    if_scan_v4<<<blocks, IF_BLOCK, 0, stream>>>(x, out, vth, n4);
  } else {
    const unsigned blocks = (n + IF_BLOCK - 1u) / IF_BLOCK;
    if_scan_scalar<<<blocks, IF_BLOCK, 0, stream>>>(x, out, vth, n);
  }
}